// GraphFilter_58780922413075
// MI455X (gfx1250) — compile-verified
//
#include <hip/hip_runtime.h>
#include <hip/hip_bf16.h>

typedef __attribute__((ext_vector_type(2))) float v2f;
typedef __attribute__((ext_vector_type(8))) float v8f;

#define F_DIM 128
#define LDS_STRIDE 132   // pad: 132 % 64 banks == 4 -> conflict-free lane spread

// ---------------------------------------------------------------------------
// Dense GEMM tile kernel: Y[16 rows x 128 cols] (+)= Z_tile @ W   (fp32 WMMA)
// 256 threads = 8 waves; wave w computes 16x16 tile at column 16*w.
// K-loop: 128 / 4 = 32 x v_wmma_f32_16x16x4_f32 per wave.
// ---------------------------------------------------------------------------
__global__ void gf_gemm_wmma(const float* __restrict__ Z,
                             const float* __restrict__ W,
                             float* __restrict__ Y, int beta) {
    __shared__ float As[16 * LDS_STRIDE];

    const int tid  = threadIdx.x;
    const int row0 = blockIdx.x * 16;

    // Stage the 16x128 A tile into LDS (contiguous 2048-float global read).
    for (int i = tid; i < 16 * F_DIM; i += 256) {
        As[(i >> 7) * LDS_STRIDE + (i & 127)] = Z[(size_t)row0 * F_DIM + i];
    }
    __syncthreads();

    const int lane = tid & 31;       // wave32
    const int wave = tid >> 5;       // 0..7
    const int hi   = lane >> 4;      // half-wave select
    const int lo   = lane & 15;
    const int c0   = wave * 16;      // output column tile

    v8f acc = {};
#pragma unroll
    for (int k = 0; k < F_DIM; k += 4) {
        const int kk = k + 2 * hi;
        // A 16x4 fragment (ISA layout): lanes0-15 hold K=0,1; lanes16-31 K=2,3
        v2f a;
        a.x = As[lo * LDS_STRIDE + kk + 0];
        a.y = As[lo * LDS_STRIDE + kk + 1];
        // B 4x16 fragment: row-major W, rows kk,kk+1; cols c0+lo (coalesced)
        v2f b;
        b.x = W[(size_t)(kk + 0) * F_DIM + c0 + lo];
        b.y = W[(size_t)(kk + 1) * F_DIM + c0 + lo];
        acc = __builtin_amdgcn_wmma_f32_16x16x4_f32(
            /*neg_a=*/false, a, /*neg_b=*/false, b,
            /*c_mod=*/(short)0, acc, /*reuse_a=*/false, /*reuse_b=*/false);
    }

    // D 16x16 layout: VGPR r -> M = r (lanes0-15) / M = 8+r (lanes16-31)
#pragma unroll
    for (int r = 0; r < 8; ++r) {
        const int row = row0 + r + 8 * hi;
        const int col = c0 + lo;
        const size_t idx = (size_t)row * F_DIM + col;
        const float v = acc[r];
        Y[idx] = beta ? (Y[idx] + v) : v;
    }
}

// Scalar tail for rows beyond the last full 16-row tile (empty for N=100000).
__global__ void gf_gemm_tail(const float* __restrict__ Z,
                             const float* __restrict__ W,
                             float* __restrict__ Y,
                             int rowStart, int nRows, int beta) {
    const int idx = blockIdx.x * blockDim.x + threadIdx.x;
    if (idx >= nRows * F_DIM) return;
    const int r = rowStart + idx / F_DIM;
    const int c = idx % F_DIM;
    float s = 0.f;
    for (int k = 0; k < F_DIM; ++k)
        s += Z[(size_t)r * F_DIM + k] * W[(size_t)k * F_DIM + c];
    const size_t o = (size_t)r * F_DIM + c;
    Y[o] = beta ? (Y[o] + s) : s;
}

// ---------------------------------------------------------------------------
// SpMM hop: Zdst[row] += (ew[e]/N) * Zsrc[col].  One wave per edge; each lane
// handles 4 floats (coalesced 512B gather; f32 atomics land in L2, z is
// L2-resident at 51 MB).
// ---------------------------------------------------------------------------
__global__ void gf_spmm_atomic(const float* __restrict__ Zsrc,
                               const int* __restrict__ ei,
                               const float* __restrict__ ew,
                               float* __restrict__ Zdst,
                               int E, float scale) {
    const long long gtid = (long long)blockIdx.x * blockDim.x + threadIdx.x;
    const int edge = (int)(gtid >> 5);
    const int lane = (int)(gtid & 31);
    if (edge >= E) return;

    const int row = ei[edge];            // edge_index[0][e]
    const int col = ei[E + edge];        // edge_index[1][e]
    const float w = ew[edge] * scale;

    const float4 v =
        *(const float4*)(Zsrc + (size_t)col * F_DIM + lane * 4);
    float* d = Zdst + (size_t)row * F_DIM + lane * 4;
    atomicAdd(d + 0, v.x * w);
    atomicAdd(d + 1, v.y * w);
    atomicAdd(d + 2, v.z * w);
    atomicAdd(d + 3, v.w * w);
}

__global__ void gf_zero(float4* __restrict__ p, long long n4) {
    const long long i = (long long)blockIdx.x * blockDim.x + threadIdx.x;
    if (i < n4) p[i] = float4{0.f, 0.f, 0.f, 0.f};
}

// ---------------------------------------------------------------------------
extern "C" void kernel_launch(void* const* d_in, const int* in_sizes, int n_in,
                              void* d_out, int out_size, void* d_ws, size_t ws_size,
                              hipStream_t stream) {
    const float* x  = (const float*)d_in[0];   // [N, 128]
    const float* ew = (const float*)d_in[1];   // [E]
    const float* Wk = (const float*)d_in[2];   // [K, 128, 128]
    const int*   ei = (const int*)d_in[3];     // [2, E]
    float*       y  = (float*)d_out;           // [N, 128]

    const int N = in_sizes[0] / F_DIM;
    const int E = in_sizes[1];
    const int K = in_sizes[2] / (F_DIM * F_DIM);
    const float scale = 1.0f / (float)N;

    float* zA = (float*)d_ws;
    float* zB = zA + (size_t)N * F_DIM;
    float* zbuf[2] = {zA, zB};

    const int nTiles = N / 16;
    const int remRows = N - nTiles * 16;

    auto run_gemm = [&](const float* Z, const float* W, int beta) {
        if (nTiles > 0)
            gf_gemm_wmma<<<nTiles, 256, 0, stream>>>(Z, W, y, beta);
        if (remRows > 0) {
            const int work = remRows * F_DIM;
            gf_gemm_tail<<<(work + 255) / 256, 256, 0, stream>>>(
                Z, W, y, nTiles * 16, remRows, beta);
        }
    };

    // k = 0: y = x @ W0  (beta=0 initializes poisoned d_out)
    run_gemm(x, Wk, 0);

    const float* zsrc = x;
    const long long n4 = (long long)N * F_DIM / 4;
    const long long spmmThreads = (long long)E * 32;

    for (int k = 1; k < K; ++k) {
        float* zdst = zbuf[(k - 1) & 1];
        gf_zero<<<(unsigned)((n4 + 255) / 256), 256, 0, stream>>>(
            (float4*)zdst, n4);
        gf_spmm_atomic<<<(unsigned)((spmmThreads + 255) / 256), 256, 0, stream>>>(
            zsrc, ei, ew, zdst, E, scale);
        run_gemm(zdst, Wk + (size_t)k * F_DIM * F_DIM, 1);
        zsrc = zdst;
    }
}